// InvariantModel_13262859010048
// MI455X (gfx1250) — compile-verified
//
#include <hip/hip_runtime.h>
#include <math.h>

namespace {

constexpr int F = 256;     // FEAT_DIM
constexpr int N = 8192;    // rows
constexpr int DEPTH = 2;

typedef __attribute__((ext_vector_type(2))) float v2f;
typedef __attribute__((ext_vector_type(8))) float v8f;

#define WMMA_F32(A, B, C) \
  __builtin_amdgcn_wmma_f32_16x16x4_f32(false, (A), false, (B), (short)0, (C), false, false)

__device__ __forceinline__ float wave_sum(float v) {
#pragma unroll
  for (int off = 16; off > 0; off >>= 1) v += __shfl_xor(v, off, 32);
  return v;
}

// a = feat_i . linear_i ; b = dirv_i . linear_i
__global__ __launch_bounds__(256) void k_scalars(const float* __restrict__ feat_i,
                                                 const float* __restrict__ dirv_i,
                                                 const float* __restrict__ lin_i,
                                                 float* __restrict__ scal) {
  __shared__ float sa[256];
  __shared__ float sb[256];
  int t = threadIdx.x;
  float l = lin_i[t];
  sa[t] = feat_i[t] * l;
  sb[t] = dirv_i[t] * l;
  __syncthreads();
  for (int s = 128; s > 0; s >>= 1) {
    if (t < s) { sa[t] += sa[t + s]; sb[t] += sb[t + s]; }
    __syncthreads();
  }
  if (t == 0) { scal[0] = sa[0]; scal[1] = sb[0]; }
}

// r[j] = sum_f E[j,f]^2   (one wave32 per row)
__global__ __launch_bounds__(256) void k_rowsq(const float* __restrict__ E,
                                               float* __restrict__ r) {
  int wave = threadIdx.x >> 5, lane = threadIdx.x & 31;
  int j = blockIdx.x * 8 + wave;
  const float4* row = reinterpret_cast<const float4*>(E + (size_t)j * F);
  float4 p = row[lane];
  float4 q = row[lane + 32];
  float v = p.x * p.x + p.y * p.y + p.z * p.z + p.w * p.w +
            q.x * q.x + q.y * q.y + q.z * q.z + q.w * q.w;
  v = wave_sum(v);
  if (lane == 0) r[j] = v;
}

// scal[2] = sum_j r[j]   (Frobenius norm^2, fixed-order reduction)
__global__ __launch_bounds__(256) void k_nrm(const float* __restrict__ r,
                                             float* __restrict__ scal) {
  __shared__ float sm[256];
  int t = threadIdx.x;
  float v = 0.f;
  for (int k = t; k < N; k += 256) v += r[k];
  sm[t] = v;
  __syncthreads();
  for (int s = 128; s > 0; s >>= 1) {
    if (t < s) sm[t] += sm[t + s];
    __syncthreads();
  }
  if (t == 0) scal[2] = sm[0];
}

// emb1[j,:] = (a - min(inner_j,0)*sign(b)/||E||) * E[j,:]
// inner_j = a*sign(b)*r[j]/||E||
__global__ __launch_bounds__(256) void k_rowscale(const float* __restrict__ E,
                                                  const float* __restrict__ r,
                                                  const float* __restrict__ scal,
                                                  float* __restrict__ O) {
  int idx = blockIdx.x * blockDim.x + threadIdx.x;  // over N*F/4 float4's
  int j = idx >> 6;                                 // F/4 = 64 per row
  float a = scal[0], b = scal[1];
  float inv_nrm = rsqrtf(scal[2]);
  float sb = copysignf(1.0f, b);
  float inner = a * sb * r[j] * inv_nrm;
  float sc = fminf(inner, 0.0f);
  float c = a - sc * sb * inv_nrm;
  float4 x = reinterpret_cast<const float4*>(E)[idx];
  x.x *= c; x.y *= c; x.z *= c; x.w *= c;
  reinterpret_cast<float4*>(O)[idx] = x;
}

// ET[f, j] = E[j, f]  (64x64 tiles through LDS, coalesced both sides)
__global__ __launch_bounds__(256) void k_transpose(const float* __restrict__ E,
                                                   float* __restrict__ ET) {
  __shared__ float lds[64][65];
  int t = threadIdx.x;
  int jBase = blockIdx.x * 64;  // grid.x = N/64
  int fBase = blockIdx.y * 64;  // grid.y = F/64
  int cg = t & 15;              // float4 group
  int jr = t >> 4;              // 0..15
#pragma unroll
  for (int p = 0; p < 4; ++p) {
    int jl = jr + p * 16;
    float4 v = *reinterpret_cast<const float4*>(
        &E[(size_t)(jBase + jl) * F + fBase + cg * 4]);
    lds[cg * 4 + 0][jl] = v.x;
    lds[cg * 4 + 1][jl] = v.y;
    lds[cg * 4 + 2][jl] = v.z;
    lds[cg * 4 + 3][jl] = v.w;
  }
  __syncthreads();
#pragma unroll
  for (int p = 0; p < 4; ++p) {
    int fl = jr + p * 16;
    float4 o;
    o.x = lds[fl][cg * 4 + 0];
    o.y = lds[fl][cg * 4 + 1];
    o.z = lds[fl][cg * 4 + 2];
    o.w = lds[fl][cg * 4 + 3];
    *reinterpret_cast<float4*>(&ET[(size_t)(fBase + fl) * N + jBase + cg * 4]) = o;
  }
}

// part[p, f] = sum over 128 rows of E[:, f]   (64 blocks)
__global__ __launch_bounds__(256) void k_colsum_part(const float* __restrict__ E,
                                                     float* __restrict__ part) {
  int f = threadIdx.x;
  int p = blockIdx.x;
  const float* base = E + (size_t)p * 128 * F + f;
  float v = 0.f;
  for (int jj = 0; jj < 128; ++jj) v += base[(size_t)jj * F];
  part[p * F + f] = v;
}

__global__ __launch_bounds__(256) void k_colsum_final(const float* __restrict__ part,
                                                      float* __restrict__ s) {
  int f = threadIdx.x;
  float v = 0.f;
  for (int p = 0; p < 64; ++p) v += part[p * F + f];
  s[f] = v;
}

// G = E^T E  via V_WMMA_F32_16X16X4_F32 on the transposed copy ET.
// Each block owns a 32x32 region of G; each wave keeps a 2x2 block of 16x16
// accumulators (A/B fragments reused twice -> 64 floats per WMMA, all b64
// loads). 8 waves split K=N; partials reduced through LDS in fixed order.
// fp32 16x4 fragment layout (ISA 7.12.2): lane L holds M(resp N)=L%16,
// K = (L>>4)*2 + vgpr.
__global__ __launch_bounds__(256) void k_gram(const float* __restrict__ ET,
                                              float* __restrict__ G) {
  int rowBase = (blockIdx.x >> 3) << 5;  // 8x8 regions of 32x32
  int colBase = (blockIdx.x & 7) << 5;
  int wave = threadIdx.x >> 5, lane = threadIdx.x & 31;
  int m = lane & 15, kh = lane >> 4;
  v8f c00 = {0.f, 0.f, 0.f, 0.f, 0.f, 0.f, 0.f, 0.f};
  v8f c01 = c00, c10 = c00, c11 = c00;
  const float* ar0 = ET + (size_t)(rowBase + m) * N;
  const float* ar1 = ET + (size_t)(rowBase + 16 + m) * N;
  const float* br0 = ET + (size_t)(colBase + m) * N;
  const float* br1 = ET + (size_t)(colBase + 16 + m) * N;
  int j0 = wave * (N / 8), j1 = j0 + (N / 8);
  for (int j = j0; j < j1; j += 4) {
    int jk = j + kh * 2;
    v2f a0 = *reinterpret_cast<const v2f*>(&ar0[jk]);
    v2f a1 = *reinterpret_cast<const v2f*>(&ar1[jk]);
    v2f b0 = *reinterpret_cast<const v2f*>(&br0[jk]);
    v2f b1 = *reinterpret_cast<const v2f*>(&br1[jk]);
    c00 = WMMA_F32(a0, b0, c00);
    c01 = WMMA_F32(a0, b1, c01);
    c10 = WMMA_F32(a1, b0, c10);
    c11 = WMMA_F32(a1, b1, c11);
  }
  __shared__ float red[4][8][32][8];  // 32 KB
#pragma unroll
  for (int v = 0; v < 8; ++v) {
    red[0][wave][lane][v] = c00[v];
    red[1][wave][lane][v] = c01[v];
    red[2][wave][lane][v] = c10[v];
    red[3][wave][lane][v] = c11[v];
  }
  __syncthreads();
  if (wave < 4) {
    int rb = rowBase + (wave >> 1) * 16;
    int cb = colBase + (wave & 1) * 16;
#pragma unroll
    for (int v = 0; v < 8; ++v) {
      float sum = 0.f;
#pragma unroll
      for (int w = 0; w < 8; ++w) sum += red[wave][w][lane][v];
      // C layout: vgpr v -> M=v (lanes 0-15) / v+8 (lanes 16-31), N=lane%16
      G[(size_t)(rb + v + 8 * kh) * F + cb + m] = sum;
    }
  }
}

// rd[j] = E[j,:] . s
__global__ __launch_bounds__(256) void k_rowdot(const float* __restrict__ E,
                                                const float* __restrict__ s,
                                                float* __restrict__ rd) {
  int wave = threadIdx.x >> 5, lane = threadIdx.x & 31;
  int j = blockIdx.x * 8 + wave;
  const float4* row = reinterpret_cast<const float4*>(E + (size_t)j * F);
  const float4* sv = reinterpret_cast<const float4*>(s);
  float4 p = row[lane], ps = sv[lane];
  float4 q = row[lane + 32], qs = sv[lane + 32];
  float v = p.x * ps.x + p.y * ps.y + p.z * ps.z + p.w * ps.w +
            q.x * qs.x + q.y * qs.y + q.z * qs.z + q.w * qs.w;
  v = wave_sum(v);
  if (lane == 0) rd[j] = v;
}

__device__ __forceinline__ void apply_tile(const float* __restrict__ E,
                                           const float* __restrict__ rd,
                                           float* __restrict__ O, v8f c,
                                           int rb, int cb, int m, int kh,
                                           float invN) {
#pragma unroll
  for (int v = 0; v < 8; ++v) {
    int row = rb + v + 8 * kh;
    int col = cb + m;
    float e = E[(size_t)row * F + col];
    O[(size_t)row * F + col] = e + (c[v] - rd[row] * e) * invN;
  }
}

// O = E + (E@G - rd .* E)/N.  256 blocks of 32 rows; wave w owns cols
// [32w, 32w+32). 2x2 register blocking; B fragments come from row-major G
// via its symmetry (G = E1^T E1 is bitwise symmetric here), so every
// fragment is a single contiguous b64 load.
__global__ __launch_bounds__(256) void k_apply(const float* __restrict__ E,
                                               const float* __restrict__ G,
                                               const float* __restrict__ rd,
                                               float* __restrict__ O) {
  const float invN = 1.0f / (float)N;
  int rowBase = blockIdx.x << 5;
  int wave = threadIdx.x >> 5, lane = threadIdx.x & 31;
  int m = lane & 15, kh = lane >> 4;
  int colBase = wave << 5;
  v8f c00 = {0.f, 0.f, 0.f, 0.f, 0.f, 0.f, 0.f, 0.f};
  v8f c01 = c00, c10 = c00, c11 = c00;
  const float* ar0 = E + (size_t)(rowBase + m) * F;
  const float* ar1 = E + (size_t)(rowBase + 16 + m) * F;
  const float* br0 = G + (size_t)(colBase + m) * F;       // G row == G col
  const float* br1 = G + (size_t)(colBase + 16 + m) * F;
  for (int k = 0; k < F; k += 4) {
    int kk = k + kh * 2;
    v2f a0 = *reinterpret_cast<const v2f*>(&ar0[kk]);
    v2f a1 = *reinterpret_cast<const v2f*>(&ar1[kk]);
    v2f b0 = *reinterpret_cast<const v2f*>(&br0[kk]);
    v2f b1 = *reinterpret_cast<const v2f*>(&br1[kk]);
    c00 = WMMA_F32(a0, b0, c00);
    c01 = WMMA_F32(a0, b1, c01);
    c10 = WMMA_F32(a1, b0, c10);
    c11 = WMMA_F32(a1, b1, c11);
  }
  apply_tile(E, rd, O, c00, rowBase, colBase, m, kh, invN);
  apply_tile(E, rd, O, c01, rowBase, colBase + 16, m, kh, invN);
  apply_tile(E, rd, O, c10, rowBase + 16, colBase, m, kh, invN);
  apply_tile(E, rd, O, c11, rowBase + 16, colBase + 16, m, kh, invN);
}

// out[j] = (E[j,:] . s)/N  for j < N-1
__global__ __launch_bounds__(256) void k_out(const float* __restrict__ E,
                                             const float* __restrict__ s,
                                             float* __restrict__ out) {
  int wave = threadIdx.x >> 5, lane = threadIdx.x & 31;
  int j = blockIdx.x * 8 + wave;
  const float4* row = reinterpret_cast<const float4*>(E + (size_t)j * F);
  const float4* sv = reinterpret_cast<const float4*>(s);
  float4 p = row[lane], ps = sv[lane];
  float4 q = row[lane + 32], qs = sv[lane + 32];
  float v = p.x * ps.x + p.y * ps.y + p.z * ps.z + p.w * ps.w +
            q.x * qs.x + q.y * qs.y + q.z * qs.z + q.w * qs.w;
  v = wave_sum(v);
  if (lane == 0 && j < N - 1) out[j] = v * (1.0f / (float)N);
}

}  // namespace

extern "C" void kernel_launch(void* const* d_in, const int* in_sizes, int n_in,
                              void* d_out, int out_size, void* d_ws,
                              size_t ws_size, hipStream_t stream) {
  (void)in_sizes; (void)n_in; (void)out_size; (void)ws_size;
  const float* X = (const float*)d_in[0];
  // d_in[1] (coefs) is unused by the reference forward path.
  const float* linear = (const float*)d_in[2];  // [DEPTH, F]
  const float* dirv = (const float*)d_in[3];    // [DEPTH, F]
  const float* feat = (const float*)d_in[4];    // [DEPTH, F]
  float* out = (float*)d_out;                   // [N-1]

  char* ws = (char*)d_ws;
  size_t off = 0;
  auto alloc = [&](size_t bytes) -> float* {
    float* p = (float*)(ws + off);
    off += (bytes + 255) & ~(size_t)255;
    return p;
  };
  float* EA = alloc((size_t)N * F * sizeof(float));   // emb1
  float* EB = alloc((size_t)N * F * sizeof(float));   // emb2 / next emb
  float* ET = alloc((size_t)N * F * sizeof(float));   // emb1 transposed
  float* G = alloc((size_t)F * F * sizeof(float));
  float* svec = alloc((size_t)F * sizeof(float));
  float* part = alloc((size_t)64 * F * sizeof(float));
  float* r = alloc((size_t)N * sizeof(float));
  float* rd = alloc((size_t)N * sizeof(float));
  float* scal = alloc(4 * sizeof(float));

  const float* cur = X;
  for (int i = 0; i < DEPTH; ++i) {
    k_scalars<<<1, 256, 0, stream>>>(feat + i * F, dirv + i * F,
                                     linear + i * F, scal);
    k_rowsq<<<N / 8, 256, 0, stream>>>(cur, r);
    k_nrm<<<1, 256, 0, stream>>>(r, scal);
    k_rowscale<<<(N * F / 4) / 256, 256, 0, stream>>>(cur, r, scal, EA);
    k_transpose<<<dim3(N / 64, F / 64), 256, 0, stream>>>(EA, ET);
    k_colsum_part<<<64, 256, 0, stream>>>(EA, part);
    k_colsum_final<<<1, 256, 0, stream>>>(part, svec);
    k_gram<<<64, 256, 0, stream>>>(ET, G);
    k_rowdot<<<N / 8, 256, 0, stream>>>(EA, svec, rd);
    k_apply<<<N / 32, 256, 0, stream>>>(EA, G, rd, EB);
    cur = EB;
  }
  // out = (emb @ colsum(emb)) / N, first N-1 rows
  k_colsum_part<<<64, 256, 0, stream>>>(EB, part);
  k_colsum_final<<<1, 256, 0, stream>>>(part, svec);
  k_out<<<N / 8, 256, 0, stream>>>(EB, svec, out);
}